// HausdorffDTLoss_28174985461969
// MI455X (gfx1250) — compile-verified
//
#include <hip/hip_runtime.h>
#include <hip/hip_bf16.h>

#define HH 384
#define WW 384
#define NIMG 8                 // B*C = 4*2
#define COLS 8                 // columns per pass-2 block
#define CHUNKS (WW / COLS)     // 48
#define NB2 (NIMG * CHUNKS)    // 384 blocks in pass 2
#define NTOT (NIMG * HH * WW)  // 1179648 elements per tensor
#define SENT 1024              // "no set bit in this row" sentinel distance

typedef float v2f __attribute__((ext_vector_type(2)));
typedef float v8f __attribute__((ext_vector_type(8)));
typedef unsigned int u32x4 __attribute__((ext_vector_type(4)));
typedef int i32x4 __attribute__((ext_vector_type(4)));
typedef int i32x8 __attribute__((ext_vector_type(8)));

#if __has_builtin(__builtin_amdgcn_tensor_load_to_lds) && \
    __has_builtin(__builtin_amdgcn_s_wait_tensorcnt)
#define HDT_HAVE_TDM 1
#endif

// ---------------------------------------------------------------------------
// Wave(32)-level sum using V_WMMA_F32_16X16X4_F32 (confirmed lowering).
// A[m][0]=acc(lane m), A[m][2]=acc(lane m+16), B=ones =>
// D[m][n] = acc[m]+acc[m+16]; summing D's 8 VGPRs leaves rows 0-7 in lanes
// 0-15 and rows 8-15 in lanes 16-31; two lane reads finish the 32-wide sum.
// ---------------------------------------------------------------------------
__device__ __forceinline__ float wave_sum32(float v) {
#if __has_builtin(__builtin_amdgcn_wmma_f32_16x16x4_f32)
  v2f a = {v, 0.0f};
  v2f b = {1.0f, 1.0f};
  v8f c = {};
  v8f d = __builtin_amdgcn_wmma_f32_16x16x4_f32(false, a, false, b, (short)0, c,
                                                false, false);
  float s = d[0] + d[1] + d[2] + d[3] + d[4] + d[5] + d[6] + d[7];
  return __shfl(s, 0, 32) + __shfl(s, 16, 32);
#else
  for (int off = 16; off > 0; off >>= 1) v += __shfl_down(v, off, 32);
  return __shfl(v, 0, 32);
#endif
}

// ---------------------------------------------------------------------------
// Pass 1: per-row 1D distance (pixels) to nearest FG / nearest BG pixel.
// One lane per (tensor,image,row); two O(W) scans; u16 output; rows with no
// set pixel keep SENT. Also sets per-image has_fg flags (idempotent atomicOr).
// Field layout in d1: f = tensor*2 + mask, mask 0 = near-FG, 1 = near-BG.
// ---------------------------------------------------------------------------
__global__ __launch_bounds__(256) void hdt_pass1_rows(
    const float* __restrict__ pred, const float* __restrict__ tgt,
    unsigned short* __restrict__ d1, unsigned int* __restrict__ flags) {
  int gid = blockIdx.x * blockDim.x + threadIdx.x;
  if (gid >= 2 * NIMG * HH) return;
  int t = gid / (NIMG * HH);
  int rem = gid - t * (NIMG * HH);
  int img = rem / HH;
  int x = rem - img * HH;

  const float* row = (t ? tgt : pred) + ((size_t)(img * HH + x)) * WW;
  unsigned short* dF = d1 + ((size_t)(((t * 2 + 0) * NIMG + img) * HH + x)) * WW;
  unsigned short* dB = d1 + ((size_t)(((t * 2 + 1) * NIMG + img) * HH + x)) * WW;

  __builtin_prefetch(row, 0, 1);  // global_prefetch_b8

  int f = SENT, b = SENT;
  bool any = false;
  for (int y = 0; y < WW; ++y) {
    bool fg = (row[y] == 1.0f);
    any |= fg;
    f = fg ? 0 : (f < SENT ? f + 1 : SENT);
    b = fg ? (b < SENT ? b + 1 : SENT) : 0;
    dF[y] = (unsigned short)f;
    dB[y] = (unsigned short)b;
  }
  f = SENT;
  b = SENT;
  for (int y = WW - 1; y >= 0; --y) {
    bool fg = (row[y] == 1.0f);
    f = fg ? 0 : (f < SENT ? f + 1 : SENT);
    b = fg ? (b < SENT ? b + 1 : SENT) : 0;
    if (f < (int)dF[y]) dF[y] = (unsigned short)f;
    if (b < (int)dB[y]) dB[y] = (unsigned short)b;
  }
  if (any) atomicOr(&flags[t * NIMG + img], 1u);
}

// ---------------------------------------------------------------------------
// Exact lower-envelope min_r( g[r] + (r-x)^2 ), early exit when d*d >= best.
// Branchless interior: indices are CLAMPED instead of skipped. Safe because a
// clamped candidate g[0]+d^2 (d>x) >= true candidate g[0]+x^2, so extra terms
// never undercut the true minimum, and all true candidates are visited before
// the exit condition can prune them.
// ---------------------------------------------------------------------------
__device__ __forceinline__ float env_min(const float* __restrict__ gc, int x) {
  float best = gc[x];
  float dd = 1.0f;
  float step = 3.0f;  // (d+1)^2 - d^2 = 2d+1
  for (int d = 1; d < HH; ++d) {
    if (dd >= best) break;
    int rm = x - d;
    int rp = x + d;
    rm = rm < 0 ? 0 : rm;
    rp = rp > (HH - 1) ? (HH - 1) : rp;
    best = fminf(best, gc[rm] + dd);
    best = fminf(best, gc[rp] + dd);
    dd += step;   // exact: integers < 2^24
    step += 2.0f;
  }
  return best;
}

// ---------------------------------------------------------------------------
// Pass 2: column parabola pass + fused loss partial reduction.
// Block = (image, 8-column chunk). Tile of 4 fields is DMA'd into LDS by the
// Tensor Data Mover (2D D#: 384 rows x 8 u16, stride 384), then converted to
// f32 squared distances (1e12 sentinel matches reference INF path exactly).
// ---------------------------------------------------------------------------
__global__ __launch_bounds__(256) void hdt_pass2_loss(
    const float* __restrict__ pred, const float* __restrict__ tgt,
    const unsigned short* __restrict__ d1, const unsigned int* __restrict__ flags,
    float* __restrict__ partials) {
  __shared__ float g[4][COLS][HH + 1];  // stride-385 f32 -> bank spread
  __shared__ float wsum[8];
#ifdef HDT_HAVE_TDM
  __shared__ unsigned short rawg[4][HH * COLS];  // TDM staging (row-major tile)
#endif

  int bid = blockIdx.x;
  int img = bid / CHUNKS;
  int y0 = (bid - img * CHUNKS) * COLS;
  int t = threadIdx.x;

  unsigned int flagP = flags[img];
  unsigned int flagT = flags[NIMG + img];

#ifdef HDT_HAVE_TDM
  // --- TDM fill: wave 0 issues one 2D tensor_load_to_lds per needed field ---
  if (t < 32) {
    for (int f = 0; f < 4; ++f) {
      if (f < 2 && !flagP) continue;  // wave-uniform skips
      if (f >= 2 && !flagT) continue;
      const unsigned short* src =
          d1 + ((size_t)(f * NIMG + img)) * HH * WW + y0;  // tile origin
      unsigned long long ga = (unsigned long long)(uintptr_t)src;
      unsigned int lds_off = (unsigned int)(uintptr_t)(&rawg[f][0]);
      u32x4 g0;
      g0[0] = 1u;                       // count=1, user mode, no gather
      g0[1] = lds_off;                  // lds_addr (bytes)
      g0[2] = (unsigned int)ga;         // global_addr[31:0]
      g0[3] = (unsigned int)(ga >> 32) & 0x01FFFFFFu;  // global_addr[56:32]
      g0[3] |= (2u << 30);              // type = 2 ("image")
      i32x8 g1;
      g1[0] = (1 << 16);                // data_size=1 (2 bytes); mask/flags 0
      g1[1] = (WW & 0xFFFF) << 16;      // tensor_dim0[15:0]
      g1[2] = (HH & 0xFFFF) << 16;      // tensor_dim0[31:16]=0 | tensor_dim1 lo
      g1[3] = (COLS << 16);             // tensor_dim1 hi = 0 | tile_dim0 = 8
      g1[4] = HH;                       // tile_dim1 = 384 | tile_dim2 = 0
      g1[5] = WW;                       // tensor_dim0_stride[31:0] = 384
      g1[6] = 0;                        // stride hi, dim1_stride lo
      g1[7] = 0;
      i32x4 z4 = {0, 0, 0, 0};          // groups 2/3 unused (2D tensor)
      i32x8 z8 = {0, 0, 0, 0, 0, 0, 0, 0};
      __builtin_amdgcn_tensor_load_to_lds(g0, g1, z4, z4, z8, 0);
    }
    __builtin_amdgcn_s_wait_tensorcnt(0);  // s_wait_tensorcnt 0
  }
  __syncthreads();
  // --- convert staging u16 -> padded f32 squared-distance arrays ---
  for (int f = 0; f < 4; ++f) {
    if (f < 2 && !flagP) continue;
    if (f >= 2 && !flagT) continue;
    for (int idx = t; idx < HH * COLS; idx += 256) {
      int col = idx % COLS;
      int r = idx / COLS;
      unsigned int dv = rawg[f][r * COLS + col];
      g[f][col][r] = (dv >= 512u) ? 1e12f : (float)(dv * dv);
    }
  }
#else
  for (int f = 0; f < 4; ++f) {
    if (f < 2 && !flagP) continue;
    if (f >= 2 && !flagT) continue;
    const unsigned short* src = d1 + ((size_t)(f * NIMG + img)) * HH * WW;
    for (int idx = t; idx < HH * COLS; idx += 256) {
      int col = idx % COLS;
      int r = idx / COLS;
      unsigned int dv = src[r * WW + y0 + col];
      g[f][col][r] = (dv >= 512u) ? 1e12f : (float)(dv * dv);
    }
  }
#endif
  __syncthreads();

  int col = t % COLS;
  int x0 = t / COLS;  // 0..31
  const float* pimg = pred + ((size_t)img) * HH * WW;
  const float* timg = tgt + ((size_t)img) * HH * WW;

  float acc = 0.0f;
  for (int k = 0; k < HH / 32; ++k) {
    int x = x0 + 32 * k;
    int y = y0 + col;
    float p2 = 0.0f;
    if (flagP) {
      float df = env_min(&g[0][col][0], x);
      float db = env_min(&g[1][col][0], x);
      float dt = sqrtf(df) + sqrtf(db);
      p2 = dt * dt;
    }
    float t2 = 0.0f;
    if (flagT) {
      float df = env_min(&g[2][col][0], x);
      float db = env_min(&g[3][col][0], x);
      float dt = sqrtf(df) + sqrtf(db);
      t2 = dt * dt;
    }
    float pv = pimg[x * WW + y];
    float tv = timg[x * WW + y];
    float e = pv - tv;
    acc += (e * e) * (p2 + t2);
  }

  // Per-wave WMMA reduction (EXEC all ones: no thread exited, flag branches
  // are block-uniform and have reconverged).
  float wtot = wave_sum32(acc);
  if ((t & 31) == 0) wsum[t >> 5] = wtot;
  __syncthreads();
  if (t == 0) {
    float s = 0.0f;
    for (int w = 0; w < 8; ++w) s += wsum[w];
    partials[bid] = s;
  }
}

__global__ __launch_bounds__(256) void hdt_finalize(
    const float* __restrict__ partials, float* __restrict__ out) {
  __shared__ float s[256];
  int t = threadIdx.x;
  float v = partials[t];
  if (t + 256 < NB2) v += partials[t + 256];
  s[t] = v;
  __syncthreads();
  for (int off = 128; off > 0; off >>= 1) {
    if (t < off) s[t] += s[t + off];
    __syncthreads();
  }
  if (t == 0) out[0] = s[0] / (float)NTOT;
}

// ---------------------------------------------------------------------------
extern "C" void kernel_launch(void* const* d_in, const int* in_sizes, int n_in,
                              void* d_out, int out_size, void* d_ws, size_t ws_size,
                              hipStream_t stream) {
  (void)in_sizes; (void)n_in; (void)out_size; (void)ws_size;
  const float* pred = (const float*)d_in[0];
  const float* tgt = (const float*)d_in[1];

  // Workspace layout:
  //   d1:       4 fields * 8 imgs * 384*384 u16  = 9,437,184 B
  //   flags:    16 u32 (has_fg per tensor-image)
  //   partials: NB2 f32
  unsigned short* d1 = (unsigned short*)d_ws;
  size_t d1_bytes = (size_t)4 * NIMG * HH * WW * sizeof(unsigned short);
  unsigned int* flags = (unsigned int*)((char*)d_ws + d1_bytes);
  float* partials = (float*)((char*)d_ws + d1_bytes + 256);

  (void)hipMemsetAsync(flags, 0, 2 * NIMG * sizeof(unsigned int), stream);

  int rows = 2 * NIMG * HH;  // 6144
  hdt_pass1_rows<<<(rows + 255) / 256, 256, 0, stream>>>(pred, tgt, d1, flags);
  hdt_pass2_loss<<<NB2, 256, 0, stream>>>(pred, tgt, d1, flags, partials);
  hdt_finalize<<<1, 256, 0, stream>>>(partials, (float*)d_out);
}